// VectorQuantizer_69320772158033
// MI455X (gfx1250) — compile-verified
//
#include <hip/hip_runtime.h>
#include <stdint.h>

typedef __attribute__((ext_vector_type(2))) float v2f;
typedef __attribute__((ext_vector_type(8))) float v8f;
typedef __attribute__((ext_vector_type(4))) unsigned int u32x4;
typedef __attribute__((ext_vector_type(8))) int i32x8;
typedef __attribute__((ext_vector_type(4))) int i32x4;

#define N_ROWS          16384   // B*H*W = 16*32*32
#define N_CODES         8192
#define CDIM            256
#define ROWS_PER_BLOCK  128     // 8 waves * 16 rows
#define SPLITS          8
#define TILES_PER_SPLIT ((N_CODES / 16) / SPLITS)   // 64 tiles of 16 codes
#define OUT_ELEMS       4194304 // 16*256*32*32
#define CROW            260     // 256 + 4 pad floats (TDM pad: interval=256dw, amount=4dw)

// ---------------------------------------------------------------- init
__global__ void vq_init(float* counts, unsigned long long* packed) {
    int i = blockIdx.x * blockDim.x + threadIdx.x;
    if (i < N_CODES) counts[i] = 0.0f;
    if (i < N_ROWS)  packed[i] = ~0ULL;
}

// ---------------------------------------------------------------- |e|^2 per code
__global__ void vq_norms(const float* __restrict__ cb, float* __restrict__ norms) {
    int tid  = threadIdx.x;
    int code = blockIdx.x * 16 + (tid >> 4);
    int l    = tid & 15;
    const float* row = cb + (size_t)code * CDIM;
    float s = 0.0f;
#pragma unroll
    for (int j = 0; j < 16; ++j) { float v = row[l + j * 16]; s += v * v; }
#pragma unroll
    for (int m = 1; m <= 8; m <<= 1) s += __shfl_xor(s, m, 32);
    if (l == 0) norms[code] = s;
}

// Issue one TDM load: 16 codes x 256 f32 tile of the codebook -> padded LDS rows.
// D# per CDNA5 ISA ch.8: group0 = {count/type, lds_addr, global_addr}, group1 =
// {data_size=4B, pad_enable, pad_interval=256dw, pad_amount=4dw, dims, tile dims, stride}.
__device__ __forceinline__ void tdm_load_tile(const float* cb, int tile, float* lds_dst) {
    unsigned long long ga = (unsigned long long)(uintptr_t)(cb + (size_t)tile * 16 * CDIM);
    unsigned ldsa = (unsigned)(uintptr_t)lds_dst;     // flat shared addr: low 32b = LDS offset

    u32x4 g0;
    g0[0] = 1u;                                       // count=1, user desc, no gather
    g0[1] = ldsa;                                     // lds_addr
    g0[2] = (unsigned)(ga & 0xFFFFFFFFull);           // global_addr[31:0]
    g0[3] = (unsigned)((ga >> 32) & 0x01FFFFFFull) | (2u << 30);  // addr[56:32] | type=2

    i32x8 g1;
    g1[0] = (int)((2u << 16) | (1u << 20) | (7u << 22) | (3u << 25)); // 4B, pad 4dw/256dw
    g1[1] = (int)(((unsigned)CDIM & 0xFFFFu) << 16);  // tensor_dim0[15:0]=256
    g1[2] = (int)(((unsigned)N_CODES & 0xFFFFu) << 16); // dim0 hi=0 | tensor_dim1 lo=8192
    g1[3] = (int)((unsigned)CDIM << 16);              // dim1 hi=0 | tile_dim0=256
    g1[4] = 16;                                       // tile_dim1=16, tile_dim2=0
    g1[5] = CDIM;                                     // tensor_dim0_stride=256
    g1[6] = 0;
    g1[7] = 0;

    i32x4 zz4 = {0, 0, 0, 0};
    i32x8 zz8 = {0, 0, 0, 0, 0, 0, 0, 0};
    // 6-arg form (clang-23 / therock-10.0 headers): (g0, g1, g2, g3, g4, cpol)
    __builtin_amdgcn_tensor_load_to_lds(g0, g1, zz4, zz4, zz8, 0);
}

// ---------------------------------------------------------------- fused GEMM + argmin (WMMA f32 + TDM)
__global__ __launch_bounds__(256, 1) void vq_argmin(
        const float* __restrict__ z, const float* __restrict__ cb,
        const float* __restrict__ norms, unsigned long long* __restrict__ packed) {
    __shared__ __align__(16) float ctile[2][16 * CROW];   // double-buffered, 2 x 16.6 KB

    const int tid  = threadIdx.x;
    const int wave = tid >> 5;
    const int lane = tid & 31;
    const int h    = lane >> 4;                 // lane half (K sub-select)
    const int lm   = lane & 15;                 // M (A) / N (B,D) index
    const int rowBase = blockIdx.x * ROWS_PER_BLOCK + wave * 16;
    const int row  = rowBase + lm;

    // z is (B=16, C=256, H=32, W=32); row n = b*1024 + (h*32+w)
    const int zb = row >> 10;
    const int zp = row & 1023;
    const float* zptr = z + (((size_t)zb) << 18) + (((size_t)(2 * h)) << 10) + zp;

    // A-matrix 16x4 f32 layout: lane<16 holds K=0,1 ; lane>=16 holds K=2,3 (per 4-wide K step)
    v2f za[64];
#pragma unroll
    for (int ks = 0; ks < 64; ++ks) {
        za[ks][0] = zptr[((size_t)(4 * ks)) << 10];
        za[ks][1] = zptr[(((size_t)(4 * ks)) << 10) + 1024];
    }

    float minv[8];
    int   mini[8];
#pragma unroll
    for (int r = 0; r < 8; ++r) { minv[r] = 3.4e38f; mini[r] = 0; }

    const int tile0 = blockIdx.y * TILES_PER_SPLIT;

    // prologue: TDM-stage first tile
    if (wave == 0) {
        tdm_load_tile(cb, tile0, &ctile[0][0]);
        __builtin_amdgcn_s_wait_tensorcnt(0);
    }
    __syncthreads();

    for (int tt = 0; tt < TILES_PER_SPLIT; ++tt) {
        const int tile = tile0 + tt;
        const int cur  = tt & 1;

        // async-prefetch next tile into the other buffer while we compute
        if (wave == 0 && tt + 1 < TILES_PER_SPLIT)
            tdm_load_tile(cb, tile + 1, &ctile[cur ^ 1][0]);

        // B row for this lane: code lm, starting at k = 2h (conflict-free: banks 4*lm+2h..+1)
        const float* bbase = &ctile[cur][lm * CROW + 2 * h];

        v8f acc = {0.f, 0.f, 0.f, 0.f, 0.f, 0.f, 0.f, 0.f};
        v2f b0 = *(const v2f*)(bbase);
        v2f b1 = *(const v2f*)(bbase + 4);
#pragma unroll
        for (int ks = 0; ks < 64; ++ks) {
            v2f bn = b0;
            if (ks + 2 < 64) bn = *(const v2f*)(bbase + 4 * (ks + 2));  // stay 2 ahead
            acc = __builtin_amdgcn_wmma_f32_16x16x4_f32(
                    false, za[ks], false, b0, (short)0, acc, false, false);
            b0 = b1;
            b1 = bn;
        }

        const float cn   = norms[tile * 16 + lm];
        const int   code = tile * 16 + lm;
#pragma unroll
        for (int r = 0; r < 8; ++r) {       // D: VGPR r -> row (r + 8h), col lm
            float score = cn - 2.0f * acc[r];
            if (score < minv[r]) { minv[r] = score; mini[r] = code; }
        }

        // next buffer must be fully landed before anyone reads it
        if (wave == 0) __builtin_amdgcn_s_wait_tensorcnt(0);
        __syncthreads();
    }

    // argmin across the 16 columns held by each half-wave, ties -> smallest idx
#pragma unroll
    for (int r = 0; r < 8; ++r) {
        float v = minv[r];
        int   i = mini[r];
#pragma unroll
        for (int m = 1; m <= 8; m <<= 1) {
            float ov = __shfl_xor(v, m, 32);
            int   oi = __shfl_xor(i, m, 32);
            if (ov < v || (ov == v && oi < i)) { v = ov; i = oi; }
        }
        if (lm == 0) {
            unsigned fk = __float_as_uint(v);
            fk = (fk & 0x80000000u) ? ~fk : (fk | 0x80000000u);  // order-preserving key
            unsigned long long pk = (((unsigned long long)fk) << 32) | (unsigned)i;
            atomicMin(packed + (rowBase + 8 * h + r), pk);       // deterministic combine
        }
    }
}

// ---------------------------------------------------------------- gather zq + per-block SSE partials
__global__ void vq_gather(const float* __restrict__ z, const float* __restrict__ cb,
                          const unsigned long long* __restrict__ packed,
                          float* __restrict__ out, float* __restrict__ partials) {
    __shared__ float red[256];
    const int tid = threadIdx.x;
    const size_t o = (size_t)blockIdx.x * 256 + tid;   // linear index into (B,C,H,W)
    const int b = (int)(o >> 18);
    const int c = (int)((o >> 10) & 255);
    const int p = (int)(o & 1023);
    const int n = (b << 10) | p;
    const int id = (int)(unsigned)(packed[n] & 0xFFFFFFFFull);
    const float zq = cb[(size_t)id * CDIM + c];
    const float zv = z[o];
    out[o] = zq;                                        // zq_st == zq in forward
    const float d = zq - zv;
    red[tid] = d * d;
    __syncthreads();
    for (int s = 128; s > 0; s >>= 1) {
        if (tid < s) red[tid] += red[tid + s];
        __syncthreads();
    }
    if (tid == 0) partials[blockIdx.x] = red[0];
}

// ---------------------------------------------------------------- counts + idx output
__global__ void vq_counts(const unsigned long long* __restrict__ packed,
                          float* __restrict__ counts, float* __restrict__ idx_out) {
    const int n = blockIdx.x * 256 + threadIdx.x;
    if (n < N_ROWS) {
        const int id = (int)(unsigned)(packed[n] & 0xFFFFFFFFull);
        atomicAdd(counts + id, 1.0f);   // integer-valued fp adds: deterministic
        idx_out[n] = (float)id;
    }
}

// ---------------------------------------------------------------- loss + perplexity
__global__ void vq_finalize(const float* __restrict__ counts,
                            const float* __restrict__ partials,
                            float* __restrict__ scal) {
    __shared__ float red[256];
    const int tid = threadIdx.x;

    float sse = 0.0f;
    for (int j = tid; j < N_ROWS; j += 256) sse += partials[j];   // fixed order
    red[tid] = sse; __syncthreads();
    for (int s = 128; s > 0; s >>= 1) { if (tid < s) red[tid] += red[tid + s]; __syncthreads(); }
    const float total_sse = red[0];
    __syncthreads();

    float e = 0.0f;
    for (int j = tid; j < N_CODES; j += 256) {
        const float pavg = counts[j] * (1.0f / (float)N_ROWS);
        e += pavg * logf(pavg + 1e-10f);
    }
    red[tid] = e; __syncthreads();
    for (int s = 128; s > 0; s >>= 1) { if (tid < s) red[tid] += red[tid + s]; __syncthreads(); }

    if (tid == 0) {
        scal[0] = 1.25f * total_sse / (float)OUT_ELEMS;  // embed + 0.25*commit (equal fwd)
        scal[1] = expf(-red[0]);
    }
}

// ---------------------------------------------------------------- launch
extern "C" void kernel_launch(void* const* d_in, const int* in_sizes, int n_in,
                              void* d_out, int out_size, void* d_ws, size_t ws_size,
                              hipStream_t stream) {
    const float* z  = (const float*)d_in[0];   // (16,256,32,32)
    const float* cb = (const float*)d_in[1];   // (8192,256)
    float* out = (float*)d_out;

    float* wsf      = (float*)d_ws;
    float* counts   = wsf;                     // 8192 f
    float* norms    = wsf + 8192;              // 8192 f
    float* partials = wsf + 16384;             // 16384 f
    unsigned long long* packed = (unsigned long long*)(wsf + 32768);  // 16384 u64

    vq_init<<<64, 256, 0, stream>>>(counts, packed);
    vq_norms<<<N_CODES / 16, 256, 0, stream>>>(cb, norms);
    dim3 grid(N_ROWS / ROWS_PER_BLOCK, SPLITS);
    vq_argmin<<<grid, 256, 0, stream>>>(z, cb, norms, packed);
    vq_gather<<<OUT_ELEMS / 256, 256, 0, stream>>>(z, cb, packed, out, partials);
    vq_counts<<<N_ROWS / 256, 256, 0, stream>>>(packed, counts, out + OUT_ELEMS + 2);
    vq_finalize<<<1, 256, 0, stream>>>(counts, partials, out + OUT_ELEMS);
}